// TopkLayer_21388937134847
// MI455X (gfx1250) — compile-verified
//
#include <hip/hip_runtime.h>
#include <stdint.h>

// ---------------------------------------------------------------------------
// Top-k magnitude masking via 3-level radix select on fabs bit patterns.
//   key = float_bits & 0x7FFFFFFF  (monotone in |x|)
//   pass1: hist of key[30:19] (4096 bins)  -> bin B1, k_rem
//   pass2: hist of key[18:9]  (1024 bins) among prefix==B1
//   pass3: hist of key[8:0]   (512 bins)  among prefix==(B1,B2)
//   threshold T = exact 31-bit key of k-th largest |x|
//   mask pass: out = (key >= T) ? x : 0   (TDM tiles -> LDS, double-buffered)
// 128MB tensor fits in 192MB L2 => passes 2..4 are L2-resident.
// ---------------------------------------------------------------------------

typedef __attribute__((ext_vector_type(4))) unsigned int u32x4;
typedef __attribute__((ext_vector_type(8))) unsigned int u32x8;

#define THREADS     256
#define HIST1_BINS  4096
#define HIST2_BINS  1024
#define HIST3_BINS  512
#define SCALAR_WORDS 16          // ws[0]=k_remaining, ws[1]=prefix/threshold
#define MTILE       4096         // floats per TDM tile (16 KB LDS per buffer)

// ---- Tensor Data Mover: 1-D tile (nelem fp32) global -> LDS ---------------
// D# group0 (128b): count=1 | lds_addr | global_addr[56:0] | type=2 ("image")
// D# group1 (256b): data_size=4B, tensor_dim0=nelem, tensor_dim1=1,
//                   tile_dim0=nelem, tile_dim1=1, dim0_stride=nelem
// Two-operand form => VADDR2/VADDR3 = NULL (tensor <= 2D).
__device__ __forceinline__ void tdm_load_tile(const float* gptr,
                                              unsigned lds_off,
                                              unsigned nelem) {
  unsigned long long ga = (unsigned long long)(uintptr_t)gptr;
  u32x4 g0;
  g0[0] = 1u;                                             // count=1, user mode
  g0[1] = lds_off;                                        // lds_addr (bytes)
  g0[2] = (unsigned)ga;                                   // global_addr[31:0]
  g0[3] = (unsigned)((ga >> 32) & 0x1ffffffULL) | (2u << 30); // [56:32] | type=2
  u32x8 g1;
  g1[0] = 2u << 16;                                       // data_size = 4 bytes
  g1[1] = (nelem & 0xffffu) << 16;                        // tensor_dim0[15:0]
  g1[2] = (nelem >> 16) | (1u << 16);                     // tensor_dim0[31:16], tensor_dim1=1
  g1[3] = (nelem & 0xffffu) << 16;                        // tile_dim0 = nelem
  g1[4] = 1u;                                             // tile_dim1 = 1
  g1[5] = nelem;                                          // tensor_dim0_stride[31:0]
  g1[6] = 0u;
  g1[7] = 0u;
  asm volatile("tensor_load_to_lds %0, %1" :: "s"(g0), "s"(g1) : "memory");
}

// ---- scratch init (d_ws is poisoned 0xAA; re-init every launch) -----------
__global__ void tk_init_kernel(unsigned* __restrict__ ws, unsigned k) {
  if (threadIdx.x == 0) { ws[0] = k; ws[1] = 0u; }
  const int nh = HIST1_BINS + HIST2_BINS + HIST3_BINS;
  for (int i = threadIdx.x; i < nh; i += blockDim.x) ws[SCALAR_WORDS + i] = 0u;
}

// ---- histogram pass: LDS-aggregated, prefix-filtered ----------------------
__global__ void tk_hist_kernel(const float* __restrict__ x, long n4,
                               unsigned* __restrict__ ghist,
                               const unsigned* __restrict__ ws,
                               int pshift, int sshift, unsigned smask) {
  __shared__ unsigned lh[HIST1_BINS];
  const int nbins = (int)smask + 1;
  for (int b = threadIdx.x; b < nbins; b += blockDim.x) lh[b] = 0u;
  __syncthreads();

  const unsigned prefix = ws[1];       // 0 on pass 1 (pshift=31 => key>>31==0)
  const float4* x4 = (const float4*)x;
  const long stride = (long)gridDim.x * blockDim.x;
  for (long i = (long)blockIdx.x * blockDim.x + threadIdx.x; i < n4; i += stride) {
    float4 v = x4[i];
    __builtin_prefetch(&x4[i + stride], 0, 0);   // global_prefetch_b8
    unsigned k0 = __float_as_uint(v.x) & 0x7fffffffu;
    unsigned k1 = __float_as_uint(v.y) & 0x7fffffffu;
    unsigned k2 = __float_as_uint(v.z) & 0x7fffffffu;
    unsigned k3 = __float_as_uint(v.w) & 0x7fffffffu;
    if ((k0 >> pshift) == prefix) atomicAdd(&lh[(k0 >> sshift) & smask], 1u);
    if ((k1 >> pshift) == prefix) atomicAdd(&lh[(k1 >> sshift) & smask], 1u);
    if ((k2 >> pshift) == prefix) atomicAdd(&lh[(k2 >> sshift) & smask], 1u);
    if ((k3 >> pshift) == prefix) atomicAdd(&lh[(k3 >> sshift) & smask], 1u);
  }
  __syncthreads();
  for (int b = threadIdx.x; b < nbins; b += blockDim.x) {
    unsigned c = lh[b];
    if (c) atomicAdd(&ghist[b], c);
  }
}

// ---- descending scan: find bin containing the k-th largest ----------------
__global__ void tk_scan_kernel(const unsigned* __restrict__ hist, int nbins,
                               unsigned* __restrict__ ws, int level_bits) {
  __shared__ unsigned psum[THREADS];
  const int t = threadIdx.x;
  const int chunk = nbins >> 8;        // nbins in {4096,1024,512} -> {16,4,2}
  unsigned s = 0;
  for (int i = 0; i < chunk; ++i) s += hist[nbins - 1 - (t * chunk + i)];
  psum[t] = s;
  __syncthreads();
  if (t == 0) {
    const unsigned k = ws[0];
    unsigned long long cum = 0;
    int c = 0;
    for (; c < THREADS; ++c) {
      if (cum + psum[c] >= k) break;
      cum += psum[c];
    }
    if (c == THREADS) c = THREADS - 1;  // defensive; cannot happen if sum>=k
    unsigned bin = 0;
    for (int i = 0; i < chunk; ++i) {
      int b = nbins - 1 - (c * chunk + i);
      unsigned h = hist[b];
      if (cum + h >= k) { bin = (unsigned)b; ws[0] = (unsigned)(k - cum); break; }
      cum += h;
    }
    ws[1] = (ws[1] << level_bits) | bin;   // accumulate threshold prefix
  }
}

// ---- final mask pass: TDM tile -> LDS (double-buffered) -> masked store ---
__global__ void tk_mask_kernel(const float* __restrict__ x,
                               float* __restrict__ out,
                               const unsigned* __restrict__ ws, long n) {
  __shared__ float tile[2][MTILE];
  const unsigned T = ws[1];            // exact 31-bit threshold key
  const long ntiles = n / MTILE;
  const bool wave0 = (threadIdx.x < 32);   // wave32: first wave of the block

  long t0 = blockIdx.x;
  if (t0 < ntiles && wave0) {
    tdm_load_tile(x + t0 * MTILE,
                  (unsigned)(uintptr_t)(void*)&tile[0][0], MTILE);
  }
  int buf = 0;
  for (long t = t0; t < ntiles; t += gridDim.x) {
    const long tn = t + gridDim.x;
    if (wave0) {
      if (tn < ntiles) {
        tdm_load_tile(x + tn * MTILE,
                      (unsigned)(uintptr_t)(void*)&tile[buf ^ 1][0], MTILE);
        __builtin_amdgcn_s_wait_tensorcnt(1);   // oldest (current buf) done
      } else {
        __builtin_amdgcn_s_wait_tensorcnt(0);
      }
    }
    __syncthreads();
    const float4* tp = (const float4*)&tile[buf][0];
    float4* op = (float4*)(out + t * MTILE);
#pragma unroll
    for (int j = 0; j < MTILE / 4 / THREADS; ++j) {
      const int idx = threadIdx.x + j * THREADS;   // coalesced b128
      float4 v = tp[idx];
      v.x = ((__float_as_uint(v.x) & 0x7fffffffu) >= T) ? v.x : 0.0f;
      v.y = ((__float_as_uint(v.y) & 0x7fffffffu) >= T) ? v.y : 0.0f;
      v.z = ((__float_as_uint(v.z) & 0x7fffffffu) >= T) ? v.z : 0.0f;
      v.w = ((__float_as_uint(v.w) & 0x7fffffffu) >= T) ? v.w : 0.0f;
      op[idx] = v;
    }
    __syncthreads();
    buf ^= 1;
  }

  // tail (n not a multiple of MTILE; not hit for 8x2048x2048)
  const long full = ntiles * MTILE;
  for (long i = full + (long)blockIdx.x * blockDim.x + threadIdx.x; i < n;
       i += (long)gridDim.x * blockDim.x) {
    float v = x[i];
    out[i] = ((__float_as_uint(v) & 0x7fffffffu) >= T) ? v : 0.0f;
  }
}

extern "C" void kernel_launch(void* const* d_in, const int* in_sizes, int n_in,
                              void* d_out, int out_size, void* d_ws, size_t ws_size,
                              hipStream_t stream) {
  const float* x = (const float*)d_in[0];
  float* out = (float*)d_out;
  const long n = (long)in_sizes[0];                 // 33,554,432
  const unsigned k = (unsigned)((double)n * 0.1);   // matches int(RATIO * n)

  unsigned* ws    = (unsigned*)d_ws;
  unsigned* hist1 = ws + SCALAR_WORDS;
  unsigned* hist2 = hist1 + HIST1_BINS;
  unsigned* hist3 = hist2 + HIST2_BINS;

  const long n4 = n / 4;
  const int hist_blocks = 2048;
  const long ntiles = n / MTILE;
  const int mask_blocks = (int)(ntiles < 1024 ? (ntiles ? ntiles : 1) : 1024);

  tk_init_kernel<<<1, THREADS, 0, stream>>>(ws, k);

  // level 1: key[30:19], 4096 bins (HBM pass; fills L2)
  tk_hist_kernel<<<hist_blocks, THREADS, 0, stream>>>(x, n4, hist1, ws, 31, 19, 0xFFFu);
  tk_scan_kernel<<<1, THREADS, 0, stream>>>(hist1, HIST1_BINS, ws, 12);

  // level 2: key[18:9], 1024 bins (L2-resident pass)
  tk_hist_kernel<<<hist_blocks, THREADS, 0, stream>>>(x, n4, hist2, ws, 19, 9, 0x3FFu);
  tk_scan_kernel<<<1, THREADS, 0, stream>>>(hist2, HIST2_BINS, ws, 10);

  // level 3: key[8:0], 512 bins (L2-resident pass) -> exact threshold in ws[1]
  tk_hist_kernel<<<hist_blocks, THREADS, 0, stream>>>(x, n4, hist3, ws, 9, 0, 0x1FFu);
  tk_scan_kernel<<<1, THREADS, 0, stream>>>(hist3, HIST3_BINS, ws, 9);

  // mask pass: TDM-fed, bandwidth-bound (read from L2, write to HBM)
  tk_mask_kernel<<<mask_blocks, THREADS, 0, stream>>>(x, out, ws, n);
}